// SeqChamferLoss_84782654423732
// MI455X (gfx1250) — compile-verified
//
#include <hip/hip_runtime.h>

typedef __attribute__((ext_vector_type(2))) float v2f;
typedef __attribute__((ext_vector_type(8))) float v8f;

#define B_SZ 4
#define C_SZ 3
#define T_SZ 16
#define N_SZ 2048
#define NTILE (N_SZ / 16)       // 128 16-wide tiles per dimension
#define THREADS 512             // 16 waves (wave32)
#define NWAVES (THREADS / 32)
#define JPW (NTILE / NWAVES)    // 8 column tiles owned per wave
#define BIGF 3.0e38f

// One workgroup per (b,t) slice. Computes
//   loss[bt] = sum_m min_n P[n,m] + sum_n min_m P[n,m]
// with P[n,m] = |x_n - y_m|^2, x = gts points, y = preds points.
__global__ __launch_bounds__(THREADS)
void chamfer_bt_kernel(const float* __restrict__ preds,
                       const float* __restrict__ gts,
                       float* __restrict__ bt_loss) {
    const int bt = blockIdx.x;          // 0..63
    const int b  = bt / T_SZ;
    const int t  = bt % T_SZ;

    // channel rows: element (b,c,t,n) at ((b*C + c)*T + t)*N + n  (contiguous in n)
    const float* x0 = gts   + ((size_t)(b * C_SZ + 0) * T_SZ + t) * N_SZ;
    const float* x1 = gts   + ((size_t)(b * C_SZ + 1) * T_SZ + t) * N_SZ;
    const float* x2 = gts   + ((size_t)(b * C_SZ + 2) * T_SZ + t) * N_SZ;
    const float* y0 = preds + ((size_t)(b * C_SZ + 0) * T_SZ + t) * N_SZ;
    const float* y1 = preds + ((size_t)(b * C_SZ + 1) * T_SZ + t) * N_SZ;
    const float* y2 = preds + ((size_t)(b * C_SZ + 2) * T_SZ + t) * N_SZ;

    __shared__ float rx[N_SZ];      // squared norms of x points
    __shared__ float ry[N_SZ];      // squared norms of y points
    __shared__ float rowmin[N_SZ];  // min over m per row n   -> loss_2
    __shared__ float colmin[N_SZ];  // min over n per col m   -> loss_1
    __shared__ float red[THREADS];

    const int tid = threadIdx.x;

    // Phase 1: squared norms + min-array init
    for (int n = tid; n < N_SZ; n += THREADS) {
        float a = x0[n], bb = x1[n], c = x2[n];
        rx[n] = a * a + bb * bb + c * c;
        float d = y0[n], e = y1[n], f = y2[n];
        ry[n] = d * d + e * e + f * f;
        rowmin[n] = BIGF;
        colmin[n] = BIGF;
    }
    __syncthreads();

    // wave index as an SGPR so per-wave work distribution is uniform/scalar
    const int wave = __builtin_amdgcn_readfirstlane(tid >> 5);
    const int lane = tid & 31;
    const int l16  = lane & 15;
    const bool hi  = lane >= 16;        // lanes 16..31 carry K=2..3 (A/B) and M+8 (D)

    // A/B-frag K channels: lo lanes -> (c0, c1), hi lanes -> (c2, 0)
    const float* xa   = hi ? x2 : x0;
    const float* ya   = hi ? y2 : y0;
    const float kmask = hi ? 0.0f : 1.0f;   // zero-pad K=3 without branching

    // Preload this wave's 8 column tiles entirely into registers:
    // B fragments (4x16 f32 each) + their ry values + colmin offsets.
    v2f   bfr[JPW];
    float rym[JPW];
    int   mcol[JPW];
#pragma unroll
    for (int k = 0; k < JPW; ++k) {
        const int m = (wave + k * NWAVES) * 16 + l16;
        mcol[k]  = m;
        bfr[k].x = ya[m];
        bfr[k].y = y1[m] * kmask;
        rym[k]   = ry[m];
    }

    for (int i = 0; i < NTILE; ++i) {
        const int n = i * 16 + l16;

        // A fragment: 16x4 f32 (row tile i of x, K padded 3->4 with zeros)
        v2f afrag;
        afrag.x = xa[n];
        afrag.y = x1[n] * kmask;

        // rx values for the 8 rows this lane owns in the D tile (M = v + 8*hi)
        float rxv[8];
        float rmin[8];
#pragma unroll
        for (int v = 0; v < 8; ++v) {
            rxv[v]  = rx[i * 16 + v + (hi ? 8 : 0)];
            rmin[v] = BIGF;
        }

        // 8 independent WMMAs over register-resident B fragments
#pragma unroll
        for (int k = 0; k < JPW; ++k) {
            v8f acc = {};
            acc = __builtin_amdgcn_wmma_f32_16x16x4_f32(
                false, afrag, false, bfr[k], (short)0, acc, false, false);

            float cmin = BIGF;
#pragma unroll
            for (int v = 0; v < 8; ++v) {
                float p = fmaf(acc[v], -2.0f, rxv[v] + rym[k]);  // rx + ry - 2*zz
                rmin[v] = fminf(rmin[v], p);
                cmin    = fminf(cmin, p);
            }
            // per-column min: lanes l16 and l16+16 both hold column mcol[k]
            __hip_atomic_fetch_min(&colmin[mcol[k]], cmin,
                                   __ATOMIC_RELAXED, __HIP_MEMORY_SCOPE_WORKGROUP);
        }

        // fold this wave's row-min contributions for row tile i
#pragma unroll
        for (int v = 0; v < 8; ++v) {
            __hip_atomic_fetch_min(&rowmin[i * 16 + v + (hi ? 8 : 0)], rmin[v],
                                   __ATOMIC_RELAXED, __HIP_MEMORY_SCOPE_WORKGROUP);
        }
    }
    __syncthreads();

    // Phase 3: sum rowmin + colmin (deterministic tree reduction)
    float partial = 0.0f;
    for (int n = tid; n < N_SZ; n += THREADS)
        partial += rowmin[n] + colmin[n];
    red[tid] = partial;
    __syncthreads();
    for (int off = THREADS / 2; off > 0; off >>= 1) {
        if (tid < off) red[tid] += red[tid + off];
        __syncthreads();
    }
    if (tid == 0) bt_loss[bt] = red[0];
}

// Mean over the 64 (b,t) losses.
__global__ __launch_bounds__(64)
void chamfer_mean_kernel(const float* __restrict__ bt_loss, float* __restrict__ out) {
    __shared__ float s[64];
    const int tid = threadIdx.x;
    s[tid] = bt_loss[tid];
    __syncthreads();
    for (int off = 32; off > 0; off >>= 1) {
        if (tid < off) s[tid] += s[tid + off];
        __syncthreads();
    }
    if (tid == 0) out[0] = s[0] * (1.0f / (B_SZ * T_SZ));
}

extern "C" void kernel_launch(void* const* d_in, const int* in_sizes, int n_in,
                              void* d_out, int out_size, void* d_ws, size_t ws_size,
                              hipStream_t stream) {
    const float* preds = (const float*)d_in[0];
    const float* gts   = (const float*)d_in[1];
    float* bt_loss     = (float*)d_ws;          // 64 floats of scratch
    float* out         = (float*)d_out;

    chamfer_bt_kernel<<<dim3(B_SZ * T_SZ), dim3(THREADS), 0, stream>>>(preds, gts, bt_loss);
    chamfer_mean_kernel<<<dim3(1), dim3(64), 0, stream>>>(bt_loss, out);
}